// ChamferLoss_62191126446336
// MI455X (gfx1250) — compile-verified
//
#include <hip/hip_runtime.h>
#include <hip/hip_bf16.h>

typedef __attribute__((ext_vector_type(2))) float v2f;
typedef __attribute__((ext_vector_type(8))) float v8f;

#define CH_B 8
#define CH_N 4096
#define CH_M 4096
#define WAVES_PER_BLOCK 8

// For each (batch, 16-row query tile): min over all db points of squared
// distance. One V_WMMA_F32_16X16X4_F32 produces a full 16x16 tile of
// (||t||^2 - 2 q.t) by folding the -2 scale and the target norm into the
// operands:  A row = (-2qx, -2qy, -2qz, 1),  B col = (tx, ty, tz, ||t||^2).
// Epilogue: sq = ||q||^2 + D;  runmin = median(0, sq, runmin)  (== clamp+min
// since runmin >= 0 invariantly) -> one v_med3_num_f32 per row.
// q: [B, nq, 3], t: [B, nt, 3], outmin: [B, nq]
__global__ __launch_bounds__(32 * WAVES_PER_BLOCK) void chamfer_min_kernel(
    const float* __restrict__ q, const float* __restrict__ t,
    float* __restrict__ outmin, int nq, int nt) {
  const int lane   = threadIdx.x & 31;
  const int wave   = threadIdx.x >> 5;
  const int lane16 = lane & 15;
  const bool hi    = lane >= 16;

  const int tiles_per_batch = nq >> 4;
  const int tile_global = blockIdx.x * WAVES_PER_BLOCK + wave;
  const int b  = tile_global / tiles_per_batch;
  const int ti = tile_global - b * tiles_per_batch;

  const float* __restrict__ qb = q + (size_t)b * nq * 3;
  const float* __restrict__ tb = t + (size_t)b * nt * 3;

  // --- A operand: 16x4, lanes 0-15 hold K={0,1}, lanes 16-31 hold K={2,3} ---
  const float* qp = qb + (size_t)(ti * 16 + lane16) * 3;
  const float qx = qp[0], qy = qp[1], qz = qp[2];
  v2f A;
  A[0] = hi ? (-2.0f * qz) : (-2.0f * qx);  // K=0 / K=2
  A[1] = hi ? 1.0f : (-2.0f * qy);          // K=1 / K=3 (K=3 multiplies ||t||^2)

  // Loop-invariant row norms ||q_row||^2: D VGPR r holds row r (+8 hi half).
  float x2v[8];
  const int rowbase = ti * 16 + (hi ? 8 : 0);
#pragma unroll
  for (int r = 0; r < 8; ++r) {
    const float* p = qb + (size_t)(rowbase + r) * 3;
    x2v[r] = p[0] * p[0] + p[1] * p[1] + p[2] * p[2];
  }

  v8f runmin;
#pragma unroll
  for (int r = 0; r < 8; ++r) runmin[r] = 3.4e38f;

  const int ntiles_t = nt >> 4;
  for (int jt = 0; jt < ntiles_t; ++jt) {
    // --- B operand: 4x16; same per-lane packing, K=3 row carries ||t||^2 ---
    const float* tp = tb + (size_t)(jt * 16 + lane16) * 3;
    const float tx = tp[0], ty = tp[1], tz = tp[2];
    const float n2 = tx * tx + ty * ty + tz * tz;  // column norm, col = lane&15
    v2f Bm;
    Bm[0] = hi ? tz : tx;   // K=0 / K=2
    Bm[1] = hi ? n2 : ty;   // K=1 / K=3

    v8f C = {};
    // D[m,n] = ||t_n||^2 - 2 q_m . t_n
    v8f D = __builtin_amdgcn_wmma_f32_16x16x4_f32(
        /*neg_a=*/false, A, /*neg_b=*/false, Bm,
        /*c_mod=*/(short)0, C, /*reuse_a=*/false, /*reuse_b=*/false);

#pragma unroll
    for (int r = 0; r < 8; ++r) {
      const float sq = x2v[r] + D[r];  // + ||q_m||^2
      // min(runmin, max(sq, 0)) == med3(0, sq, runmin) given runmin >= 0
      runmin[r] = __builtin_amdgcn_fmed3f(0.0f, sq, runmin[r]);
    }
  }

  // Min across the 16 columns held by each 16-lane half (wave32: xor masks
  // 1,2,4,8 stay inside each half).
#pragma unroll
  for (int r = 0; r < 8; ++r) {
    float v = runmin[r];
#pragma unroll
    for (int m = 1; m < 16; m <<= 1) v = fminf(v, __shfl_xor(v, m, 32));
    runmin[r] = v;
  }

  if (lane16 == 0) {  // lane 0 writes rows 0-7, lane 16 writes rows 8-15
#pragma unroll
    for (int r = 0; r < 8; ++r)
      outmin[(size_t)b * nq + rowbase + r] = runmin[r];
  }
}

// Single-block reduction: out = mean(min1) + mean(min2)
__global__ __launch_bounds__(256) void chamfer_reduce_kernel(
    const float* __restrict__ mins, float* __restrict__ out, int n1, int n2) {
  __shared__ float s1[256];
  __shared__ float s2[256];
  float a = 0.0f, c = 0.0f;
  for (int i = threadIdx.x; i < n1; i += 256) a += mins[i];
  for (int i = threadIdx.x; i < n2; i += 256) c += mins[n1 + i];
  s1[threadIdx.x] = a;
  s2[threadIdx.x] = c;
  __syncthreads();
  for (int s = 128; s > 0; s >>= 1) {
    if (threadIdx.x < s) {
      s1[threadIdx.x] += s1[threadIdx.x + s];
      s2[threadIdx.x] += s2[threadIdx.x + s];
    }
    __syncthreads();
  }
  if (threadIdx.x == 0)
    out[0] = s1[0] / (float)n1 + s2[0] / (float)n2;
}

extern "C" void kernel_launch(void* const* d_in, const int* in_sizes, int n_in,
                              void* d_out, int out_size, void* d_ws, size_t ws_size,
                              hipStream_t stream) {
  const float* input  = (const float*)d_in[0];   // [B, N, 3]
  const float* target = (const float*)d_in[1];   // [B, M, 3]
  float* out = (float*)d_out;

  float* min1 = (float*)d_ws;                    // [B, N]
  float* min2 = min1 + (size_t)CH_B * CH_N;      // [B, M]

  const int blocks1 = (CH_B * (CH_N / 16)) / WAVES_PER_BLOCK;
  const int blocks2 = (CH_B * (CH_M / 16)) / WAVES_PER_BLOCK;

  // Pass 1: nearest target for each input point
  chamfer_min_kernel<<<blocks1, 32 * WAVES_PER_BLOCK, 0, stream>>>(
      input, target, min1, CH_N, CH_M);
  // Pass 2: nearest input for each target point
  chamfer_min_kernel<<<blocks2, 32 * WAVES_PER_BLOCK, 0, stream>>>(
      target, input, min2, CH_M, CH_N);
  // Final: mean(min1) + mean(min2)
  chamfer_reduce_kernel<<<1, 256, 0, stream>>>(min1, out, CH_B * CH_N, CH_B * CH_M);
}